// S6_40389872451937
// MI455X (gfx1250) — compile-verified
//
#include <hip/hip_runtime.h>

// ---------------------------------------------------------------------------
// S6 (Mamba) forward for MI455X / gfx1250.
//   B=2, L=1024, Dm=1024, Di=2048, N=16 (all compile-time).
// GEMMs: V_WMMA_F32_16X16X4_F32, one 16x(JT*16) tile per wave, no divergent
// control flow in the main loop (EXEC stays all-ones through every WMMA),
// software-pipelined global loads.
// Selective scan: chunk-parallel linear recurrence (8 chunks of 128).
// ---------------------------------------------------------------------------

typedef __attribute__((ext_vector_type(2))) float v2f;
typedef __attribute__((ext_vector_type(8))) float v8f;

#define BATCH  2
#define SEQ    1024
#define DM     1024
#define DI     2048
#define NST    16
#define ML     (BATCH * SEQ)      // 2048 GEMM rows
#define CHUNKS 8
#define TC     (SEQ / CHUNKS)     // 128
#define NCH    (BATCH * DI)       // 4096 scan channels

// ---------------------------------------------------------------------------
// NT GEMM:  C[m,n] = act( sum_k A[m,k] * W[n,k] + bias[n] )
// A: (M,K) row-major, W: (N,K) row-major, C: (M,ldc).
// Template JT: number of 16-column tiles per wave (tile = 16 x JT*16).
// Template K: reduction depth (compile-time -> immediate offsets).
// Grid is flat: global tile id = blockIdx.x*8 + wave; tiles_n = 1<<ln2_tn.
// Every wave owns a full tile -> no guards, no exec manipulation.
// ---------------------------------------------------------------------------
template <int JT, int K>
__global__ __launch_bounds__(256) void wmma_gemm_nt(
    const float* __restrict__ A, const float* __restrict__ W,
    float* __restrict__ C, int ldc, int ln2_tn,
    const float* __restrict__ bias, int act)
{
    const int lane = threadIdx.x & 31;
    const int wave = threadIdx.x >> 5;
    const int gt   = blockIdx.x * 8 + wave;          // global tile id
    const int m0   = (gt >> ln2_tn) << 4;
    const int n0   = (gt & ((1 << ln2_tn) - 1)) * (JT * 16);

    const int ml   = lane & 15;    // row (A) / col (B,C,D) within tile
    const int kh   = lane >> 4;    // lane-half -> K pair (A,B) / +8 rows (C,D)
    const int kofs = 2 * kh;

    const float* __restrict__ arow  = A + (size_t)(m0 + ml) * K + kofs;
    const float* __restrict__ wbase = W + (size_t)(n0 + ml) * K + kofs;

    v8f acc[JT];
#pragma unroll
    for (int j = 0; j < JT; ++j)
#pragma unroll
        for (int r = 0; r < 8; ++r) acc[j][r] = 0.0f;

    // -------- software-pipelined K loop (peeled prologue/epilogue) --------
    v2f a = *(const v2f*)(arow);
    v2f b[JT];
#pragma unroll
    for (int j = 0; j < JT; ++j)
        b[j] = *(const v2f*)(wbase + (size_t)j * 16 * K);

    for (int k0 = 4; k0 < K; k0 += 4) {
        // issue next K-step loads first: they overlap the WMMAs below
        v2f a2 = *(const v2f*)(arow + k0);
        v2f b2[JT];
#pragma unroll
        for (int j = 0; j < JT; ++j)
            b2[j] = *(const v2f*)(wbase + (size_t)j * 16 * K + k0);
#pragma unroll
        for (int j = 0; j < JT; ++j)
            acc[j] = __builtin_amdgcn_wmma_f32_16x16x4_f32(
                false, a, false, b[j], (short)0, acc[j], false, false);
        a = a2;
#pragma unroll
        for (int j = 0; j < JT; ++j) b[j] = b2[j];
    }
#pragma unroll
    for (int j = 0; j < JT; ++j)
        acc[j] = __builtin_amdgcn_wmma_f32_16x16x4_f32(
            false, a, false, b[j], (short)0, acc[j], false, false);

    // C/D layout: VGPR r -> M = m0 + r (lanes 0-15) / m0 + 8 + r (lanes 16-31)
#pragma unroll
    for (int j = 0; j < JT; ++j) {
        const int n = n0 + j * 16 + ml;
        const float bn = (bias != nullptr) ? bias[n] : 0.0f;
#pragma unroll
        for (int r = 0; r < 8; ++r) {
            const int m = m0 + r + (kh << 3);
            float v = acc[j][r] + bn;
            if (act == 1) v = (v > 20.0f) ? v : log1pf(__expf(v));  // softplus
            C[(size_t)m * ldc + n] = v;
        }
    }
}

// ---------------------------------------------------------------------------
// Depthwise conv1d (k=3, pad=1) over L, + bias, + SiLU.
// Reads xs half of xz (B,L,2*DI); writes xs_act (B,L,DI).
// ---------------------------------------------------------------------------
__global__ __launch_bounds__(256) void conv_silu(
    const float* __restrict__ xz, const float* __restrict__ cw,
    const float* __restrict__ cb, float* __restrict__ xs_act)
{
    const int idx = blockIdx.x * blockDim.x + threadIdx.x;  // 0 .. ML*DI-1
    const int d   = idx & (DI - 1);
    const int row = idx >> 11;          // b*SEQ + t
    const int t   = row & (SEQ - 1);
    const float* base = xz + (size_t)row * (2 * DI) + d;
    const float w0 = cw[d * 3 + 0], w1 = cw[d * 3 + 1], w2 = cw[d * 3 + 2];
    const float xm = (t > 0)       ? base[-(ptrdiff_t)(2 * DI)] : 0.0f;
    const float xc = base[0];
    const float xp = (t < SEQ - 1) ? base[2 * DI] : 0.0f;
    float v = fmaf(w0, xm, fmaf(w1, xc, fmaf(w2, xp, cb[d])));
    xs_act[idx] = v / (1.0f + __expf(-v));                  // SiLU
}

// ---------------------------------------------------------------------------
// Scan pass 1: per (chunk, b, d) compute per-state decay product P[n] and
// local recurrence h[n] starting from 0.  idx = c*NCH + ch (coalesced on d).
// ---------------------------------------------------------------------------
__global__ __launch_bounds__(256) void scan_pass1(
    const float* __restrict__ delta, const float* __restrict__ xs,
    const float* __restrict__ bc, const float* __restrict__ A_log,
    float* __restrict__ Psum, float* __restrict__ Hloc)
{
    const int idx = blockIdx.x * blockDim.x + threadIdx.x;
    const int ch  = idx & (NCH - 1);
    const int c   = idx >> 12;
    const int b   = ch >> 11;
    const int d   = ch & (DI - 1);

    float An[NST], P[NST], h[NST];
#pragma unroll
    for (int n = 0; n < NST; ++n) {
        An[n] = -__expf(A_log[d * NST + n]);   // A = -exp(A_log)
        P[n] = 1.0f;
        h[n] = 0.0f;
    }

    const int t0 = c * TC;
    for (int t = t0; t < t0 + TC; ++t) {
        const int row = b * SEQ + t;
        const float dt = delta[(size_t)row * DI + d];
        const float xv = xs[(size_t)row * DI + d];
        const float dx = dt * xv;
        const float* __restrict__ Bv = bc + (size_t)row * (2 * NST);
#pragma unroll
        for (int n = 0; n < NST; ++n) {
            const float dA = __expf(dt * An[n]);
            P[n] *= dA;
            h[n] = fmaf(dA, h[n], dx * Bv[n]);
        }
    }
    const size_t off = (size_t)idx * NST;
#pragma unroll
    for (int n = 0; n < NST; ++n) { Psum[off + n] = P[n]; Hloc[off + n] = h[n]; }
}

// ---------------------------------------------------------------------------
// Scan pass 2: serial combine of the 8 chunk summaries per channel; emits the
// true initial state of each chunk.
// ---------------------------------------------------------------------------
__global__ __launch_bounds__(256) void scan_pass2(
    const float* __restrict__ Psum, const float* __restrict__ Hloc,
    float* __restrict__ Hinit)
{
    const int ch = blockIdx.x * blockDim.x + threadIdx.x;   // 0..NCH-1
    float hi[NST];
#pragma unroll
    for (int n = 0; n < NST; ++n) hi[n] = 0.0f;
    for (int c = 0; c < CHUNKS; ++c) {
        const size_t off = ((size_t)c * NCH + ch) * NST;
#pragma unroll
        for (int n = 0; n < NST; ++n) {
            Hinit[off + n] = hi[n];
            hi[n] = fmaf(Psum[off + n], hi[n], Hloc[off + n]);
        }
    }
}

// ---------------------------------------------------------------------------
// Scan pass 3: replay each chunk from its true initial state; stream hidden
// states (B,L,DI,NST) with float4 stores (64B contiguous per thread) and
// produce the gated output y2 = (y + xs*D) * silu(z).
// ---------------------------------------------------------------------------
__global__ __launch_bounds__(256) void scan_pass3(
    const float* __restrict__ delta, const float* __restrict__ xs,
    const float* __restrict__ bc, const float* __restrict__ xz,
    const float* __restrict__ A_log, const float* __restrict__ Dvec,
    const float* __restrict__ Hinit, float* __restrict__ y2,
    float* __restrict__ hidden)
{
    const int idx = blockIdx.x * blockDim.x + threadIdx.x;
    const int ch  = idx & (NCH - 1);
    const int c   = idx >> 12;
    const int b   = ch >> 11;
    const int d   = ch & (DI - 1);

    float An[NST], h[NST];
    const size_t ioff = (size_t)idx * NST;
#pragma unroll
    for (int n = 0; n < NST; ++n) {
        An[n] = -__expf(A_log[d * NST + n]);
        h[n] = Hinit[ioff + n];
    }
    const float Dd = Dvec[d];

    const int t0 = c * TC;
    for (int t = t0; t < t0 + TC; ++t) {
        const int row = b * SEQ + t;
        const float dt = delta[(size_t)row * DI + d];
        const float xv = xs[(size_t)row * DI + d];
        const float dx = dt * xv;
        const float* __restrict__ Bv = bc + (size_t)row * (2 * NST);
        const float* __restrict__ Cv = Bv + NST;
        float y = 0.0f;
#pragma unroll
        for (int n = 0; n < NST; ++n) {
            const float dA = __expf(dt * An[n]);
            h[n] = fmaf(dA, h[n], dx * Bv[n]);
            y = fmaf(h[n], Cv[n], y);
        }
        float4* __restrict__ hout =
            (float4*)(hidden + ((size_t)row * DI + d) * NST);
#pragma unroll
        for (int q = 0; q < 4; ++q)
            hout[q] = make_float4(h[4 * q], h[4 * q + 1], h[4 * q + 2], h[4 * q + 3]);

        const float zv = xz[(size_t)row * (2 * DI) + DI + d];
        const float sz = zv / (1.0f + __expf(-zv));         // silu(z)
        y2[(size_t)row * DI + d] = (y + xv * Dd) * sz;
    }
}

// ---------------------------------------------------------------------------
extern "C" void kernel_launch(void* const* d_in, const int* in_sizes, int n_in,
                              void* d_out, int out_size, void* d_ws, size_t ws_size,
                              hipStream_t stream)
{
    const float* x       = (const float*)d_in[0];   // (B,L,DM)
    const float* w_in    = (const float*)d_in[1];   // (2*DI, DM)
    const float* conv_w  = (const float*)d_in[2];   // (DI,1,3)
    const float* conv_b  = (const float*)d_in[3];   // (DI)
    const float* w_xproj = (const float*)d_in[4];   // (2*NST, DI)
    const float* w_dt    = (const float*)d_in[5];   // (DI, DI)
    const float* b_dt    = (const float*)d_in[6];   // (DI)
    const float* w_out   = (const float*)d_in[7];   // (DM, DI)
    const float* A_log   = (const float*)d_in[8];   // (DI, NST)
    const float* Dvec    = (const float*)d_in[9];   // (DI)

    float* out    = (float*)d_out;                  // (B,L,DM)
    float* hidden = out + (size_t)ML * DM;          // (B,L,DI,NST)

    // Workspace carve-up (~91 MB of f32)
    float* xz     = (float*)d_ws;                       // ML * 2*DI
    float* xs_act = xz     + (size_t)ML * 2 * DI;       // ML * DI
    float* bc     = xs_act + (size_t)ML * DI;           // ML * 2*NST
    float* delta  = bc     + (size_t)ML * 2 * NST;      // ML * DI
    float* y2     = delta  + (size_t)ML * DI;           // ML * DI
    float* Psum   = y2     + (size_t)ML * DI;           // CHUNKS*NCH*NST
    float* Hloc   = Psum   + (size_t)CHUNKS * NCH * NST;
    float* Hinit  = Hloc   + (size_t)CHUNKS * NCH * NST;

    const dim3 blk(256);

    // Tile bookkeeping: tile = 16 x (JT*16); tiles = (M/16) * (N/(JT*16));
    // blocks = tiles / 8 waves.  ln2_tn = log2(N / (JT*16)).
    // 1) in_proj: xz = x @ w_in^T        M=2048, N=4096, K=1024, JT=8
    //    tiles_n = 32 (ln2 5), tiles = 128*32 = 4096 -> 512 blocks
    wmma_gemm_nt<8, DM><<<dim3(512), blk, 0, stream>>>(
        x, w_in, xz, 2 * DI, 5, nullptr, 0);

    // 2) depthwise conv + SiLU on xs half
    conv_silu<<<dim3((ML * DI) / 256), blk, 0, stream>>>(xz, conv_w, conv_b, xs_act);

    // 3) x_proj: BC = xs_act @ w_xproj^T  M=2048, N=32, K=2048, JT=2
    //    tiles_n = 1 (ln2 0), tiles = 128 -> 16 blocks
    wmma_gemm_nt<2, DI><<<dim3(16), blk, 0, stream>>>(
        xs_act, w_xproj, bc, 2 * NST, 0, nullptr, 0);

    // 4) dt_proj + bias + softplus: delta M=2048, N=2048, K=2048, JT=8
    //    tiles_n = 16 (ln2 4), tiles = 128*16 = 2048 -> 256 blocks
    wmma_gemm_nt<8, DI><<<dim3(256), blk, 0, stream>>>(
        xs_act, w_dt, delta, DI, 4, b_dt, 1);

    // 5) chunked selective scan
    scan_pass1<<<dim3((CHUNKS * NCH) / 256), blk, 0, stream>>>(
        delta, xs_act, bc, A_log, Psum, Hloc);
    scan_pass2<<<dim3(NCH / 256), blk, 0, stream>>>(Psum, Hloc, Hinit);
    scan_pass3<<<dim3((CHUNKS * NCH) / 256), blk, 0, stream>>>(
        delta, xs_act, bc, xz, A_log, Dvec, Hinit, y2, hidden);

    // 6) out_proj: out = y2 @ w_out^T     M=2048, N=1024, K=2048, JT=8
    //    tiles_n = 8 (ln2 3), tiles = 128*8 = 1024 -> 128 blocks
    wmma_gemm_nt<8, DI><<<dim3(128), blk, 0, stream>>>(
        y2, w_out, out, DM, 3, nullptr, 0);
}